// GCNEncoder_36017595744478
// MI455X (gfx1250) — compile-verified
//
#include <hip/hip_runtime.h>
#include <hip/hip_bf16.h>

typedef float v2f __attribute__((ext_vector_type(2)));
typedef float v4f __attribute__((ext_vector_type(4)));
typedef float v8f __attribute__((ext_vector_type(8)));

#define GCN_IN  128
#define GCN_HID 128
#define GCN_OUT 64
#define GCN_BN_EPS 1e-5f

#if defined(__HIP_DEVICE_COMPILE__) && defined(__gfx1250__) && __has_builtin(__builtin_amdgcn_wmma_f32_16x16x4_f32)
#define HAVE_WMMA_F32 1
#else
#define HAVE_WMMA_F32 0
#endif

// ---------------------------------------------------------------------------
// Degree / normalization
// ---------------------------------------------------------------------------
__global__ void gcn_init_deg(float* __restrict__ deg, int n) {
    int i = blockIdx.x * blockDim.x + threadIdx.x;
    if (i < n) deg[i] = 1.0f;  // self-loop contributes 1 to every node
}

__global__ void gcn_deg_accum(const int* __restrict__ dst, float* __restrict__ deg, int nE) {
    int e = blockIdx.x * blockDim.x + threadIdx.x;
    if (e < nE) atomicAdd(&deg[dst[e]], 1.0f);
}

__global__ void gcn_dinv(const float* __restrict__ deg, float* __restrict__ dinv, int n) {
    int i = blockIdx.x * blockDim.x + threadIdx.x;
    if (i < n) {
        float d = deg[i];
        dinv[i] = (d > 0.0f) ? rsqrtf(d) : 0.0f;
    }
}

// ---------------------------------------------------------------------------
// GEMM: S[m][c] = (sum_k X[m][k] * W[k][c]) * dinv[m];  T = S (self-loop init)
// One wave computes a 16-row x NCOL strip via V_WMMA_F32_16X16X4_F32.
// Block = 256 threads = 8 waves = 128 rows.
// W staged in LDS K-pair interleaved: Wp[p][n] = {W[2p][n], W[2p+1][n]},
// so each B fragment is one contiguous ds_load_b64 into an even VGPR pair.
// ---------------------------------------------------------------------------
template <int NCOL>
__global__ __launch_bounds__(256)
void gcn_gemm_scale(const float* __restrict__ X, const float* __restrict__ W,
                    const float* __restrict__ dinv,
                    float* __restrict__ S, float* __restrict__ T, int n)
{
    __shared__ float Wp[GCN_IN * NCOL];   // interleaved pairs, 64KB max
    const int tid = threadIdx.x;
    for (int pn = tid; pn < (GCN_IN / 2) * NCOL; pn += 256) {
        int p  = pn >> (NCOL == 128 ? 7 : 6);
        int nn = pn & (NCOL - 1);
        v2f w;
        w.x = W[(2 * p + 0) * NCOL + nn];
        w.y = W[(2 * p + 1) * NCOL + nn];
        *(v2f*)&Wp[pn * 2] = w;
    }
    __syncthreads();

    const int wave = tid >> 5;
    const int lane = tid & 31;
    const int half = lane >> 4;   // 0: lanes 0-15, 1: lanes 16-31
    const int lr   = lane & 15;

    const int m0 = blockIdx.x * 128 + wave * 16;

    constexpr int NT = NCOL / 16;
    v8f acc[NT];
#pragma unroll
    for (int t = 0; t < NT; ++t)
        acc[t] = v8f{0.f, 0.f, 0.f, 0.f, 0.f, 0.f, 0.f, 0.f};

#if HAVE_WMMA_F32
    // A fragment (16x4 f32): lane lr holds row m0+lr; VGPR pair holds
    // K = {half*2, half*2+1} relative to the k-step base.
    const int arow = min(m0 + lr, n - 1);   // clamp: keep EXEC all-ones
    const float* __restrict__ xrow = X + (long)arow * GCN_IN;
    for (int k = 0; k < GCN_IN; k += 4) {
        v2f a = *(const v2f*)&xrow[k + half * 2];
        const int p = (k >> 1) + half;      // pair index: covers K = {2p, 2p+1}
#pragma unroll
        for (int t = 0; t < NT; ++t) {
            v2f b = *(const v2f*)&Wp[(p * NCOL + t * 16 + lr) * 2];
            acc[t] = __builtin_amdgcn_wmma_f32_16x16x4_f32(
                false, a, false, b, (short)0, acc[t], false, false);
        }
    }
#else
    // Scalar fallback in the identical C/D lane layout (safety net only).
    for (int k = 0; k < GCN_IN; ++k) {
#pragma unroll
        for (int t = 0; t < NT; ++t) {
            float bv = Wp[((k >> 1) * NCOL + t * 16 + lr) * 2 + (k & 1)];
#pragma unroll
            for (int r = 0; r < 8; ++r) {
                int row = min(m0 + half * 8 + r, n - 1);
                acc[t][r] += X[(long)row * GCN_IN + k] * bv;
            }
        }
    }
#endif

    // C/D layout: VGPR r -> row m0 + half*8 + r, col t*16 + lr
#pragma unroll
    for (int t = 0; t < NT; ++t) {
#pragma unroll
        for (int r = 0; r < 8; ++r) {
            int row = m0 + half * 8 + r;
            if (row < n) {
                float v = acc[t][r] * dinv[row];
                long idx = (long)row * NCOL + t * 16 + lr;
                S[idx] = v;
                T[idx] = v;   // self-loop term: T starts as S
            }
        }
    }
}

// ---------------------------------------------------------------------------
// Edge scatter: T[dst] += S[src]
// One thread per (edge, 16B chunk): b128 gather + 4 f32 atomics (L2-resident).
// ---------------------------------------------------------------------------
template <int NCOL>
__global__ void gcn_scatter(const int* __restrict__ src, const int* __restrict__ dst,
                            const float* __restrict__ S, float* __restrict__ T, int nE)
{
    constexpr int CH = NCOL / 4;          // 16B chunks per row
    long idx = (long)blockIdx.x * blockDim.x + threadIdx.x;
    long total = (long)nE * CH;
    if (idx >= total) return;
    int e = (int)(idx >> (NCOL == 128 ? 5 : 4));
    int q = (int)(idx & (CH - 1));
    int s = src[e];
    int d = dst[e];
    v4f v = *(const v4f*)&S[(long)s * NCOL + q * 4];
    float* tp = &T[(long)d * NCOL + q * 4];
    atomicAdd(tp + 0, v.x);
    atomicAdd(tp + 1, v.y);
    atomicAdd(tp + 2, v.z);
    atomicAdd(tp + 3, v.w);
}

// ---------------------------------------------------------------------------
// H = dinv[row]*T + bias;  accumulate per-channel sum / sumsq into stats[0..2*HID)
// Block handles 32 rows x 128 channels; LDS tree then channel atomics.
// ---------------------------------------------------------------------------
__global__ void gcn_zero_stats(float* __restrict__ stats) {
    int i = threadIdx.x;
    if (i < 2 * GCN_HID) stats[i] = 0.0f;
}

__global__ __launch_bounds__(256)
void gcn_finish_stats(const float* __restrict__ T, const float* __restrict__ dinv,
                      const float* __restrict__ bias, float* __restrict__ H,
                      float* __restrict__ stats, int n)
{
    __shared__ float sh[512];
    const int tid = threadIdx.x;
    const int c   = tid & (GCN_HID - 1);
    const int rg  = tid >> 7;             // 0 or 1
    const int row0 = blockIdx.x * 32;
    const float b = bias[c];
    float s = 0.0f, sq = 0.0f;
    for (int r = rg; r < 32; r += 2) {
        int row = row0 + r;
        if (row < n) {
            float h = dinv[row] * T[(long)row * GCN_HID + c] + b;
            H[(long)row * GCN_HID + c] = h;
            s += h;
            sq += h * h;
        }
    }
    sh[tid]       = s;
    sh[256 + tid] = sq;
    __syncthreads();
    if (tid < 128) {
        atomicAdd(&stats[c],           sh[tid] + sh[tid + 128]);
        atomicAdd(&stats[GCN_HID + c], sh[256 + tid] + sh[256 + tid + 128]);
    }
}

// stats -> scale/shift in place: scale = g*rsqrt(var+eps), shift = b - mean*scale
__global__ void gcn_bn_finalize(float* __restrict__ stats, const float* __restrict__ gamma,
                                const float* __restrict__ beta, int n)
{
    int c = threadIdx.x;
    if (c < GCN_HID) {
        float inv_n = 1.0f / (float)n;
        float m = stats[c] * inv_n;
        float v = stats[GCN_HID + c] * inv_n - m * m;   // biased variance
        float sc = gamma[c] * rsqrtf(v + GCN_BN_EPS);
        stats[c] = sc;
        stats[GCN_HID + c] = beta[c] - m * sc;
    }
}

// Xout = relu(H*scale + shift) + Xres   (vectorized b128)
__global__ void gcn_bn_relu_res(const float* __restrict__ H, const float* __restrict__ stats,
                                const float* __restrict__ Xres, float* __restrict__ Xout, int n)
{
    long i4 = (long)blockIdx.x * blockDim.x + threadIdx.x;
    if (i4 >= (long)n * (GCN_HID / 4)) return;
    long base = i4 * 4;
    int c0 = (int)(base & (GCN_HID - 1));
    v4f h  = *(const v4f*)&H[base];
    v4f sc = *(const v4f*)&stats[c0];
    v4f sf = *(const v4f*)&stats[GCN_HID + c0];
    v4f xr = *(const v4f*)&Xres[base];
    v4f y;
#pragma unroll
    for (int j = 0; j < 4; ++j)
        y[j] = fmaxf(fmaf(h[j], sc[j], sf[j]), 0.0f) + xr[j];
    *(v4f*)&Xout[base] = y;
}

// out = dinv[row]*T + b3   (final conv, no BN; vectorized b128)
__global__ void gcn_final_out(const float* __restrict__ T, const float* __restrict__ dinv,
                              const float* __restrict__ bias, float* __restrict__ out, int n)
{
    long i4 = (long)blockIdx.x * blockDim.x + threadIdx.x;
    if (i4 >= (long)n * (GCN_OUT / 4)) return;
    long base = i4 * 4;
    int row = (int)(base >> 6);
    int c0  = (int)(base & (GCN_OUT - 1));
    float dv = dinv[row];
    v4f t = *(const v4f*)&T[base];
    v4f b = *(const v4f*)&bias[c0];
    v4f y;
#pragma unroll
    for (int j = 0; j < 4; ++j)
        y[j] = fmaf(dv, t[j], b[j]);
    *(v4f*)&out[base] = y;
}

// ---------------------------------------------------------------------------
static inline int cdiv(long a, long b) { return (int)((a + b - 1) / b); }

extern "C" void kernel_launch(void* const* d_in, const int* in_sizes, int n_in,
                              void* d_out, int out_size, void* d_ws, size_t ws_size,
                              hipStream_t stream) {
    (void)n_in; (void)out_size; (void)ws_size;

    const float* x    = (const float*)d_in[0];
    const int*   ei   = (const int*)d_in[1];
    const float* W1   = (const float*)d_in[2];
    const float* b1   = (const float*)d_in[3];
    const float* W2   = (const float*)d_in[4];
    const float* b2   = (const float*)d_in[5];
    const float* W3   = (const float*)d_in[6];
    const float* b3   = (const float*)d_in[7];
    const float* g1   = (const float*)d_in[8];
    const float* be1  = (const float*)d_in[9];
    const float* g2   = (const float*)d_in[10];
    const float* be2  = (const float*)d_in[11];

    const int N = in_sizes[0] / GCN_IN;
    const int E = in_sizes[1] / 2;
    const int* srcE = ei;          // edge_index[0]
    const int* dstE = ei + E;      // edge_index[1]

    // workspace layout (floats): B0, B1, B2 (N*HID each), deg, dinv, stats
    float* ws = (float*)d_ws;
    size_t nh = (size_t)N * GCN_HID;
    float* B0    = ws;
    float* B1    = B0 + nh;
    float* B2    = B1 + nh;
    float* deg   = B2 + nh;
    float* dinv  = deg + N;
    float* stats = dinv + N;

    const int TPB = 256;

    // --- normalization coefficients -------------------------------------
    gcn_init_deg<<<cdiv(N, TPB), TPB, 0, stream>>>(deg, N);
    gcn_deg_accum<<<cdiv(E, TPB), TPB, 0, stream>>>(dstE, deg, E);
    gcn_dinv<<<cdiv(N, TPB), TPB, 0, stream>>>(deg, dinv, N);

    const int gemmBlocks = cdiv(N, 128);

    // --- layer 1 ---------------------------------------------------------
    gcn_gemm_scale<GCN_HID><<<gemmBlocks, TPB, 0, stream>>>(x, W1, dinv, B0, B1, N);
    gcn_scatter<GCN_HID><<<cdiv((long)E * (GCN_HID / 4), TPB), TPB, 0, stream>>>(srcE, dstE, B0, B1, E);
    gcn_zero_stats<<<1, TPB, 0, stream>>>(stats);
    gcn_finish_stats<<<cdiv(N, 32), TPB, 0, stream>>>(B1, dinv, b1, B0, stats, N);
    gcn_bn_finalize<<<1, GCN_HID, 0, stream>>>(stats, g1, be1, N);
    gcn_bn_relu_res<<<cdiv((long)N * (GCN_HID / 4), TPB), TPB, 0, stream>>>(B0, stats, x, B2, N);

    // --- layer 2 ---------------------------------------------------------
    gcn_gemm_scale<GCN_HID><<<gemmBlocks, TPB, 0, stream>>>(B2, W2, dinv, B0, B1, N);
    gcn_scatter<GCN_HID><<<cdiv((long)E * (GCN_HID / 4), TPB), TPB, 0, stream>>>(srcE, dstE, B0, B1, E);
    gcn_zero_stats<<<1, TPB, 0, stream>>>(stats);
    gcn_finish_stats<<<cdiv(N, 32), TPB, 0, stream>>>(B1, dinv, b2, B0, stats, N);
    gcn_bn_finalize<<<1, GCN_HID, 0, stream>>>(stats, g2, be2, N);
    gcn_bn_relu_res<<<cdiv((long)N * (GCN_HID / 4), TPB), TPB, 0, stream>>>(B0, stats, B2, B1, N);

    // --- output layer (HID -> OUT, no BN) --------------------------------
    gcn_gemm_scale<GCN_OUT><<<gemmBlocks, TPB, 0, stream>>>(B1, W3, dinv, B0, B2, N);
    gcn_scatter<GCN_OUT><<<cdiv((long)E * (GCN_OUT / 4), TPB), TPB, 0, stream>>>(srcE, dstE, B0, B2, E);
    gcn_final_out<<<cdiv((long)N * (GCN_OUT / 4), TPB), TPB, 0, stream>>>(B2, dinv, b3, (float*)d_out, N);
}